// SolidHarmonics_34127810134283
// MI455X (gfx1250) — compile-verified
//
#include <hip/hip_runtime.h>
#include <cstdint>

#define LMAX   8
#define BLK    256
#define STAGEP 257   // staging column stride in floats (bank-spread for 64 banks)

typedef float v4f __attribute__((ext_vector_type(4)));

// ---------------- compile-time coefficients ----------------
struct Coefs {
    float F[LMAX + 1][LMAX + 1];   // (-1)^m sqrt((2l+1)/(2pi) (l-m)!/(l+m)!)
    float K0[LMAX + 1];            // sqrt((2l+1)/(4pi))
    float All[LMAX + 1];           // Q[l][l] = (-1)^l (2l-1)!!
    float Bl[LMAX + 1];            // Q[l][l-1] = Bl * z
    float Ar[LMAX + 1][LMAX + 1];  // (2l-1)/(l-m)
    float Br[LMAX + 1][LMAX + 1];  // (l+m-1)/(l-m)
};

constexpr double cfact(int n) { double r = 1.0; for (int i = 2; i <= n; ++i) r *= (double)i; return r; }
constexpr double csqrt_(double x) {
    if (x <= 0.0) return 0.0;
    double g = x < 1.0 ? 1.0 : x;
    for (int i = 0; i < 256; ++i) g = 0.5 * (g + x / g);
    return g;
}
constexpr Coefs make_coefs() {
    Coefs C{};
    const double PI = 3.14159265358979323846264338327950288;
    double all = 1.0;
    for (int l = 0; l <= LMAX; ++l) {
        if (l > 0) all *= -(2.0 * l - 1.0);
        C.All[l] = (float)all;
    }
    {
        double prev = 1.0;
        for (int l = 1; l <= LMAX; ++l) {
            C.Bl[l] = (float)((2.0 * l - 1.0) * prev);
            prev *= -(2.0 * l - 1.0);
        }
    }
    for (int l = 0; l <= LMAX; ++l) {
        C.K0[l] = (float)csqrt_((2.0 * l + 1.0) / (4.0 * PI));
        for (int m = 1; m <= l; ++m) {
            double v = csqrt_((2.0 * l + 1.0) / (2.0 * PI) * cfact(l - m) / cfact(l + m));
            C.F[l][m] = (float)(((m & 1) ? -1.0 : 1.0) * v);
        }
        for (int m = 0; m <= l - 2; ++m) {
            C.Ar[l][m] = (float)((2.0 * l - 1.0) / (double)(l - m));
            C.Br[l][m] = (float)((double)(l + m - 1) / (double)(l - m));
        }
    }
    return C;
}
__constant__ const Coefs CF = make_coefs();

// ---------------- gfx1250 async global->LDS input stage ----------------
#if defined(__HIP_DEVICE_COMPILE__) && defined(__gfx1250__) && \
    __has_builtin(__builtin_amdgcn_global_load_async_to_lds_b32)
#define USE_ASYNC_LDS 1
#else
#define USE_ASYNC_LDS 0
#endif

__device__ __forceinline__ void stage_inputs(const float* __restrict__ R, float* s_in,
                                             int base3, int tid, int n3) {
#if USE_ASYNC_LDS
#pragma unroll
    for (int r = 0; r < 3; ++r) {
        const int idx = r * BLK + tid;
        if (base3 + idx < n3) {
            __builtin_amdgcn_global_load_async_to_lds_b32(
                (__attribute__((address_space(1))) int*)(R + base3 + idx),
                (__attribute__((address_space(3))) int*)(s_in + idx),
                0, 0);
        } else {
            s_in[idx] = 0.0f;
        }
    }
    asm volatile("s_wait_asynccnt 0" ::: "memory");
#else
#pragma unroll
    for (int r = 0; r < 3; ++r) {
        const int idx = r * BLK + tid;
        s_in[idx] = (base3 + idx < n3) ? R[base3 + idx] : 0.0f;
    }
#endif
}

// ---------------- kernel ----------------
__global__ __launch_bounds__(BLK) void solid_harmonics_kernel(const float* __restrict__ R,
                                                              float* __restrict__ out,
                                                              int N) {
    __shared__ float s_in[3 * BLK];
    __shared__ float s_stage[(2 * LMAX + 1) * STAGEP];

    const int tid = threadIdx.x;
    const int p0  = blockIdx.x * BLK;

    // coalesced (async on gfx1250) input stage: 768 contiguous floats per block
    stage_inputs(R, s_in, p0 * 3, tid, 3 * N);
    __syncthreads();

    const float x = s_in[3 * tid + 0];
    const float y = s_in[3 * tid + 1];
    const float z = s_in[3 * tid + 2];
    const float r2 = x * x + y * y + z * z;

    // c_m + i s_m = (x + i y)^m
    float cc[LMAX + 1], ss[LMAX + 1];
    cc[0] = 1.0f; ss[0] = 0.0f;
#pragma unroll
    for (int m = 1; m <= LMAX; ++m) {
        cc[m] = x * cc[m - 1] - y * ss[m - 1];
        ss[m] = x * ss[m - 1] + y * cc[m - 1];
    }

    const int npts = (N - p0) < BLK ? (N - p0) : BLK;

    float Qp[LMAX + 1];   // row l-1
    float Qpp[LMAX + 1];  // row l-2

#pragma unroll
    for (int l = 0; l <= LMAX; ++l) {
        // ---- Q row l ----
        float Qc[LMAX + 1];
        if (l == 0) {
            Qc[0] = 1.0f;
        } else {
            Qc[l]     = CF.All[l];
            Qc[l - 1] = CF.Bl[l] * z;
#pragma unroll
            for (int m = l - 2; m >= 0; --m)
                Qc[m] = CF.Ar[l][m] * z * Qp[m] - CF.Br[l][m] * r2 * Qpp[m];
        }

        // ---- stage the 2l+1 columns (column-major in LDS: [col][point]) ----
        const int W = 2 * l + 1;
#pragma unroll
        for (int j = 0; j < W; ++j) {
            float v;
            if (j < l) {                       // sin terms, m = l..1
                const int m = l - j;
                v = CF.F[l][m] * Qc[m] * ss[m];
            } else if (j == l) {               // m = 0
                v = CF.K0[l] * Qc[0];
            } else {                           // cos terms, m = 1..l
                const int m = j - l;
                v = CF.F[l][m] * Qc[m] * cc[m];
            }
            s_stage[j * STAGEP + tid] = v;
        }
        __syncthreads();

        // ---- cooperative, coalesced streaming store of this l-segment ----
        {
            const size_t segBase = (size_t)N * (size_t)(l * l);
            const size_t blkOff  = (size_t)p0 * (size_t)W;
            float* __restrict__ segp = out + segBase + blkOff;
            const int total = npts * W;

            if (((total & 3) == 0) && (((segBase | blkOff) & 3) == 0)) {
                const int nq = total >> 2;  // float4 count
                for (int q = tid; q < nq; q += BLK) {
                    const int k  = q << 2;
                    const int i0 = (k + 0) / W, c0 = (k + 0) - i0 * W;
                    const int i1 = (k + 1) / W, c1 = (k + 1) - i1 * W;
                    const int i2 = (k + 2) / W, c2 = (k + 2) - i2 * W;
                    const int i3 = (k + 3) / W, c3 = (k + 3) - i3 * W;
                    v4f v;
                    v.x = s_stage[c0 * STAGEP + i0];
                    v.y = s_stage[c1 * STAGEP + i1];
                    v.z = s_stage[c2 * STAGEP + i2];
                    v.w = s_stage[c3 * STAGEP + i3];
                    __builtin_nontemporal_store(v, reinterpret_cast<v4f*>(segp + k));
                }
            } else {  // generic fallback (not hit for N = 1e6)
                for (int k = tid; k < total; k += BLK) {
                    const int i = k / W, c = k - i * W;
                    __builtin_nontemporal_store(s_stage[c * STAGEP + i], segp + k);
                }
            }
        }
        __syncthreads();

        // ---- rotate Q rows ----
#pragma unroll
        for (int m = 0; m <= l; ++m) { Qpp[m] = Qp[m]; Qp[m] = Qc[m]; }
    }
}

// ---------------- launch ----------------
extern "C" void kernel_launch(void* const* d_in, const int* in_sizes, int n_in,
                              void* d_out, int out_size, void* d_ws, size_t ws_size,
                              hipStream_t stream) {
    const float* R = (const float*)d_in[0];
    float* out = (float*)d_out;
    const int N = in_sizes[0] / 3;
    const int blocks = (N + BLK - 1) / BLK;
    hipLaunchKernelGGL(solid_harmonics_kernel, dim3(blocks), dim3(BLK), 0, stream, R, out, N);
}